// feat_prototype_distance_module_43422119363105
// MI455X (gfx1250) — compile-verified
//
#include <hip/hip_runtime.h>

typedef __attribute__((ext_vector_type(2))) float v2f;
typedef __attribute__((ext_vector_type(8))) float v8f;

#define C_DIM 256
#define HW    32768          // 128*256
#define N_IMG 4
#define K_CLS 19
#define KPAD  32

__global__ __launch_bounds__(256) void feat_proto_dist_wmma(
    const float* __restrict__ feat,     // [N, C, H, W]
    const float* __restrict__ protos,   // [1, K, C] -> flat k*C + c
    float* __restrict__ out)            // [N, K, H, W]
{
  __shared__ float lds_protos[KPAD * C_DIM];   // 32 KB, zero-padded rows 19..31
  __shared__ float lds_p2[KPAD];

  const int tid = threadIdx.x;

  // Cooperative load of prototypes into LDS, zero-pad k >= 19.
  for (int i = tid; i < KPAD * C_DIM; i += 256) {
    int k = i >> 8;                       // C_DIM == 256
    lds_protos[i] = (k < K_CLS) ? protos[i] : 0.0f;
  }
  __syncthreads();

  // p2[k] = sum_c protos[k,c]^2
  if (tid < KPAD) {
    const float* row = &lds_protos[tid * C_DIM];
    float s = 0.0f;
    for (int c = 0; c < C_DIM; ++c) s += row[c] * row[c];
    lds_p2[tid] = s;
  }
  __syncthreads();

  const int wave = tid >> 5;
  const int lane = tid & 31;
  const int half = lane >> 4;    // 0: lanes 0-15, 1: lanes 16-31
  const int lx   = lane & 15;

  // Each wave handles a 16-column strip of the flattened (n, hw) space.
  const int  strip = blockIdx.x * 8 + wave;
  const long m0    = (long)strip * 16;
  const int  n     = (int)(m0 / HW);
  const int  hw    = (int)(m0 % HW) + lx;

  // B operand: lane supplies feat[c0 + 2*half + r][hw] for VGPR r (r=0,1)
  const float* fptr  = feat + ((size_t)n * C_DIM + 2 * half) * (size_t)HW + hw;
  // A operand: lane supplies protos[lx][c0 + 2*half + r] -> one b64 LDS load
  const float* aptr  = &lds_protos[lx * C_DIM + 2 * half];
  const float* aptrh = aptr + 16 * C_DIM;

  v8f acc_lo = {};
  v8f acc_hi = {};
  float f2p = 0.0f;   // partial ||f||^2 (this lane's c mod 4 residues)

  #pragma unroll 4
  for (int c0 = 0; c0 < C_DIM; c0 += 4) {
    float bx = fptr[0];
    float by = fptr[HW];
    fptr += 4 * (size_t)HW;

    v2f b; b[0] = bx; b[1] = by;
    f2p += bx * bx + by * by;

    v2f a_lo = *(const v2f*)(aptr  + c0);   // 8B-aligned ds_load_b64
    v2f a_hi = *(const v2f*)(aptrh + c0);

    // 8 args: (neg_a, A, neg_b, B, c_mod, C, reuse_a, reuse_b)
    acc_lo = __builtin_amdgcn_wmma_f32_16x16x4_f32(
        false, a_lo, false, b, (short)0, acc_lo, false, false);
    acc_hi = __builtin_amdgcn_wmma_f32_16x16x4_f32(
        false, a_hi, false, b, (short)0, acc_hi, false, false);
  }

  // Complete ||f||^2: partner half-wave (lane ^ 16) holds the other residues.
  float f2 = f2p + __shfl_xor(f2p, 16, 32);

  // C/D layout: VGPR r, this lane -> (k_row = r + 8*half, col = lx)
  const size_t outbase = (size_t)n * K_CLS * HW + hw;
  #pragma unroll
  for (int r = 0; r < 8; ++r) {
    const int k_lo = r + 8 * half;
    float d2 = f2 + lds_p2[k_lo] - 2.0f * acc_lo[r];
    d2 = fmaxf(d2, 0.0f);
    out[outbase + (size_t)k_lo * HW] = sqrtf(d2);

    const int k_hi = 16 + k_lo;
    if (k_hi < K_CLS) {
      float e2 = f2 + lds_p2[k_hi] - 2.0f * acc_hi[r];
      e2 = fmaxf(e2, 0.0f);
      out[outbase + (size_t)k_hi * HW] = sqrtf(e2);
    }
  }
}

extern "C" void kernel_launch(void* const* d_in, const int* in_sizes, int n_in,
                              void* d_out, int out_size, void* d_ws, size_t ws_size,
                              hipStream_t stream) {
  const float* feat   = (const float*)d_in[0];
  const float* protos = (const float*)d_in[1];
  // d_in[2] is class_numbers (==19), fixed by the problem; hardcoded as K_CLS.
  float* out = (float*)d_out;

  const int total_strips = (N_IMG * HW) / 16;   // 8192
  const int blocks = total_strips / 8;          // 1024 blocks x 8 waves
  feat_proto_dist_wmma<<<blocks, 256, 0, stream>>>(feat, protos, out);
}